// VectorQuantizer_46213848105138
// MI455X (gfx1250) — compile-verified
//
#include <hip/hip_runtime.h>
#include <math.h>

typedef float v2f __attribute__((ext_vector_type(2)));
typedef float v8f __attribute__((ext_vector_type(8)));
typedef unsigned int u32x4 __attribute__((ext_vector_type(4)));
typedef int i32x4 __attribute__((ext_vector_type(4)));
typedef int i32x8 __attribute__((ext_vector_type(8)));

#define K_EMB  1024
#define D_EMB  64
#define N_ROWS 131072            // 32 * 64 * 64
#define TK     128               // embedding rows staged in LDS per chunk
#define LSTR   66                // padded LDS row stride (floats); TDM pad: 2 dwords / 64 dwords
#define WAVES  8                 // 256-thread workgroup = 8 wave32

// ---------------------------------------------------------------------------
// Kernel 1: embedding squared norms + zero scratch (deterministic each call)
// ---------------------------------------------------------------------------
__global__ __launch_bounds__(256) void vq_prep(const float* __restrict__ emb,
                                               float* __restrict__ e2,
                                               int* __restrict__ counts,
                                               float* __restrict__ sqsum) {
    int k = blockIdx.x * blockDim.x + threadIdx.x;
    if (k < K_EMB) {
        float s = 0.f;
        #pragma unroll
        for (int d = 0; d < D_EMB; ++d) {
            float v = emb[(size_t)k * D_EMB + d];
            s = fmaf(v, v, s);
        }
        e2[k] = s;
        counts[k] = 0;
    }
    if (k == 0) *sqsum = 0.f;
}

// ---------------------------------------------------------------------------
// Kernel 2: fused distance-matmul (fp32 WMMA) + argmin + gather + losses
//   latents: [32, 64, 64, 64] (B, D, H, W); row n = b*4096 + h*64 + w
//   element (n, d) lives at ((n>>12)*64 + d)*4096 + (n & 4095)
//   Embedding chunks staged to LDS via the Tensor Data Mover (TDM), using
//   D# padding (2 dwords every 64) to realize the bank-conflict-free stride 66.
// ---------------------------------------------------------------------------
__global__ __launch_bounds__(256) void vq_main(const float* __restrict__ lat,
                                               const float* __restrict__ emb,
                                               const float* __restrict__ e2g,
                                               float* __restrict__ outq,
                                               int* __restrict__ inds,
                                               int* __restrict__ counts,
                                               float* __restrict__ sqsum) {
    __shared__ float ldsE[TK * LSTR];
    __shared__ float ldsE2[TK];
    __shared__ int   indbuf[WAVES * 16];

    const int tid  = threadIdx.x;
    const int wv   = tid >> 5;
    const int lane = tid & 31;
    const int lh   = lane >> 4;      // lane half (selects K-pair in A/B layouts)
    const int ll   = lane & 15;      // row (A) / col (B) within tile

    const int rowBase = (blockIdx.x * WAVES + wv) * 16;

    // ---- A tile: 16 rows x 64 depth, ISA 16x4 f32 layout per K-step ----
    const int    nA    = rowBase + ll;
    const size_t rbase = ((size_t)(nA >> 12) * D_EMB) * 4096 + (size_t)(nA & 4095);
    v2f aT[16];
    #pragma unroll
    for (int kk = 0; kk < 16; ++kk) {
        int d0 = 4 * kk + 2 * lh;
        aT[kk].x = lat[rbase + (size_t)d0 * 4096];
        aT[kk].y = lat[rbase + (size_t)(d0 + 1) * 4096];
    }

    float minv[8];
    int   mini[8];
    #pragma unroll
    for (int j = 0; j < 8; ++j) { minv[j] = 3.4e38f; mini[j] = 0; }

    const uint32_t lds_base = (uint32_t)(uintptr_t)&ldsE[0];

    for (int kb = 0; kb < K_EMB; kb += TK) {
        __syncthreads();           // previous chunk fully consumed

        if (wv == 0) {
            // --- TDM descriptor: 128x64 f32 tile, global -> LDS, padded ---
            uint64_t ga = (uint64_t)(uintptr_t)(emb + (size_t)kb * D_EMB);
            u32x4 g0;
            g0[0] = 1u;                                  // count=1 (valid, user)
            g0[1] = lds_base;                            // lds_addr (bytes)
            g0[2] = (uint32_t)(ga & 0xFFFFFFFFu);        // global_addr[31:0]
            g0[3] = (uint32_t)((ga >> 32) & 0x1FFFFFFu)  // global_addr[56:32]
                  | (2u << 30);                          // type = 2 (image)
            i32x8 g1;
            g1[0] = (int)((2u << 16)      // data_size = 4 bytes
                        | (1u << 20)      // pad_enable
                        | (5u << 22)      // pad_interval: every 64 dwords
                        | (1u << 25));    // pad_amount: 2 dwords -> stride 66
            g1[1] = (int)(64u  << 16);    // tensor_dim0 = 64   (bits [63:48])
            g1[2] = (int)(128u << 16);    // tensor_dim1 = 128  (bits [95:80])
            g1[3] = (int)(64u  << 16);    // tile_dim0   = 64   (bits [127:112])
            g1[4] = 128;                  // tile_dim1 = 128; tile_dim2 = 0
            g1[5] = 64;                   // tensor_dim0_stride = 64
            g1[6] = 0;
            g1[7] = 0;
            i32x4 gz4 = {0, 0, 0, 0};
            i32x8 gz8 = {0, 0, 0, 0, 0, 0, 0, 0};
            __builtin_amdgcn_tensor_load_to_lds(g0, g1, gz4, gz4, gz8, 0);
            __builtin_amdgcn_s_wait_tensorcnt(0);
        }
        for (int i = tid; i < TK; i += 256) ldsE2[i] = e2g[kb + i];
        __syncthreads();           // chunk (TDM) + norms (DS) visible to all

        #pragma unroll
        for (int ct = 0; ct < TK / 16; ++ct) {
            v8f acc = {};
            const float* bp = &ldsE[(ct * 16 + ll) * LSTR + 2 * lh];
            #pragma unroll
            for (int kk = 0; kk < 16; ++kk) {
                v2f bv = *(const v2f*)(bp + 4 * kk);
                acc = __builtin_amdgcn_wmma_f32_16x16x4_f32(
                    false, aT[kk], false, bv, (short)0, acc, false, false);
            }
            const int   col = kb + ct * 16 + ll;
            const float e2v = ldsE2[ct * 16 + ll];
            #pragma unroll
            for (int j = 0; j < 8; ++j) {
                float dist = fmaf(-2.f, acc[j], e2v);   // ||e||^2 - 2 f.e
                if (dist < minv[j]) { minv[j] = dist; mini[j] = col; }
            }
        }
    }

    // min-reduce across the 16 lanes holding the same row (xor masks stay in-half)
    #pragma unroll
    for (int m = 1; m <= 8; m <<= 1) {
        #pragma unroll
        for (int j = 0; j < 8; ++j) {
            float ov = __shfl_xor(minv[j], m, 32);
            int   oi = __shfl_xor(mini[j], m, 32);
            if (ov < minv[j] || (ov == minv[j] && oi < mini[j])) {
                minv[j] = ov; mini[j] = oi;
            }
        }
    }

    if (ll == 0) {                 // lane 0 -> rows 0..7, lane 16 -> rows 8..15
        #pragma unroll
        for (int j = 0; j < 8; ++j) {
            int r = j + 8 * lh;
            indbuf[wv * 16 + r] = mini[j];
            inds[rowBase + r]   = mini[j];
            atomicAdd(&counts[mini[j]], 1);
        }
    }
    __syncthreads();

    // gather winning embedding, write quantized output (==straight-through fwd),
    // accumulate squared error
    float sq = 0.f;
    const int    n2     = rowBase + ll;
    const size_t rbase2 = ((size_t)(n2 >> 12) * D_EMB) * 4096 + (size_t)(n2 & 4095);
    const int    myind  = indbuf[wv * 16 + ll];
    #pragma unroll
    for (int i = 0; i < 32; ++i) {
        int   d  = 2 * i + lh;
        float q  = emb[(size_t)myind * D_EMB + d];
        float lv = lat[rbase2 + (size_t)d * 4096];
        float df = q - lv;
        sq = fmaf(df, df, sq);
        outq[rbase2 + (size_t)d * 4096] = q;
    }
    #pragma unroll
    for (int m = 1; m < 32; m <<= 1) sq += __shfl_xor(sq, m, 32);
    if (lane == 0) atomicAdd(sqsum, sq);
}

// ---------------------------------------------------------------------------
// Kernel 3: one_hot [N, K] fp32 stream (512 MB) — the true bandwidth limiter
// ---------------------------------------------------------------------------
__global__ __launch_bounds__(256) void vq_onehot(const int* __restrict__ inds,
                                                 float* __restrict__ oh) {
    size_t t = (size_t)blockIdx.x * blockDim.x + threadIdx.x;  // N*K/4 threads
    int nrow = (int)(t >> 8);
    int k0   = (int)(t & 255) * 4;
    int ind  = inds[nrow];
    v2f lo, hi;
    lo.x = (k0     == ind) ? 1.f : 0.f;
    lo.y = (k0 + 1 == ind) ? 1.f : 0.f;
    hi.x = (k0 + 2 == ind) ? 1.f : 0.f;
    hi.y = (k0 + 3 == ind) ? 1.f : 0.f;
    // output region is 8B-aligned (offset 8388610 floats) -> two b64 stores
    *(v2f*)(oh + t * 4)     = lo;
    *(v2f*)(oh + t * 4 + 2) = hi;
}

// ---------------------------------------------------------------------------
// Kernel 4: scalars — vq_loss = 1.25 * MSE ; perplexity from histogram
// ---------------------------------------------------------------------------
__global__ __launch_bounds__(1024) void vq_final(const int* __restrict__ counts,
                                                 const float* __restrict__ sqsum,
                                                 float* __restrict__ out01) {
    __shared__ float red[1024];
    int k = threadIdx.x;
    float p = (float)counts[k] * (1.0f / (float)N_ROWS);
    red[k] = p * logf(p + 1e-10f);
    __syncthreads();
    for (int s = 512; s > 0; s >>= 1) {
        if (k < s) red[k] += red[k + s];
        __syncthreads();
    }
    if (k == 0) {
        out01[1] = expf(-red[0]);
        out01[0] = (*sqsum) * (1.25f / (float)(N_ROWS * D_EMB));
    }
}

// ---------------------------------------------------------------------------
extern "C" void kernel_launch(void* const* d_in, const int* in_sizes, int n_in,
                              void* d_out, int out_size, void* d_ws, size_t ws_size,
                              hipStream_t stream) {
    const float* lat = (const float*)d_in[0];   // [32,64,64,64] fp32
    const float* emb = (const float*)d_in[1];   // [1024,64] fp32

    float* out  = (float*)d_out;
    float* outq = out + 2;                                  // quantized [B,D,H,W]
    float* oh   = out + 2 + (size_t)N_ROWS * D_EMB;         // one_hot [N,K]

    char*  ws     = (char*)d_ws;
    int*   inds   = (int*)ws;                               // N ints
    int*   counts = (int*)(ws + (size_t)N_ROWS * 4);        // K ints
    float* sqsum  = (float*)(ws + (size_t)N_ROWS * 4 + 4096);
    float* e2     = (float*)(ws + (size_t)N_ROWS * 4 + 4096 + 256);

    vq_prep<<<4, 256, 0, stream>>>(emb, e2, counts, sqsum);
    vq_main<<<N_ROWS / (16 * WAVES), 256, 0, stream>>>(lat, emb, e2, outq,
                                                       inds, counts, sqsum);
    vq_onehot<<<(N_ROWS / 256) * (K_EMB / 4), 256, 0, stream>>>(inds, oh);
    vq_final<<<1, 1024, 0, stream>>>(counts, sqsum, out);
}